// GCNConv_1065151889890
// MI455X (gfx1250) — compile-verified
//
#include <hip/hip_runtime.h>
#include <hip/hip_bf16.h>

typedef __attribute__((ext_vector_type(2))) float v2f;
typedef __attribute__((ext_vector_type(8))) float v8f;
typedef unsigned int v4u __attribute__((ext_vector_type(4)));
typedef int v4i __attribute__((ext_vector_type(4)));
typedef int v8i __attribute__((ext_vector_type(8)));

#define N_NODES   10000
#define D_FEAT    256
#define ROW_WORDS 313                                   // ceil(10000 / 32)
#define MASK_BYTES ((size_t)N_NODES * ROW_WORDS * 4)    // 12,520,000 (16B aligned)
#define APAD 260                                        // 256 + 4 dword pad per row

// ---------------------------------------------------------------------------
// K0: zero the adjacency bitmask (grid-stride uint4 fill)
// ---------------------------------------------------------------------------
__global__ void gcn_zero(uint4* __restrict__ p, long n16) {
  long i = (long)blockIdx.x * blockDim.x + threadIdx.x;
  const long stride = (long)gridDim.x * blockDim.x;
  const uint4 z = {0u, 0u, 0u, 0u};
  for (; i < n16; i += stride) p[i] = z;
}

// ---------------------------------------------------------------------------
// K1: symmetric binarized adjacency bitmask via atomicOr
//     (matches jnp at[src,dst].max(ones): duplicate edges collapse to 1)
// ---------------------------------------------------------------------------
__global__ void gcn_build_adj(const int* __restrict__ ei, int E,
                              unsigned* __restrict__ mask) {
  const int e = blockIdx.x * blockDim.x + threadIdx.x;
  if (e >= E) return;
  const int s = ei[e];       // edge_index row 0
  const int d = ei[E + e];   // edge_index row 1
  atomicOr(&mask[(size_t)s * ROW_WORDS + (d >> 5)], 1u << (d & 31));
  atomicOr(&mask[(size_t)d * ROW_WORDS + (s >> 5)], 1u << (s & 31));
}

// ---------------------------------------------------------------------------
// K2: deg[i] = 1 (+I) + popcount(row i);  dinv = 1/sqrt(deg)
// ---------------------------------------------------------------------------
__global__ void gcn_degree(const unsigned* __restrict__ mask,
                           float* __restrict__ dinv) {
  const int i = blockIdx.x * blockDim.x + threadIdx.x;
  if (i >= N_NODES) return;
  const unsigned* row = mask + (size_t)i * ROW_WORDS;
  int cnt = 1;  // diagonal from +I
  for (int w = 0; w < ROW_WORDS; ++w) cnt += __popc(row[w]);
  dinv[i] = rsqrtf((float)cnt);
}

// ---------------------------------------------------------------------------
// K3: h[i,:] = dinv[i] * ( dinv[i]*x[i,:] + sum_{j in row(i)} dinv[j]*x[j,:] )
//     Block per node, feature per lane; deterministic ascending-j scan.
//     x (10.24 MB) is fully L2-resident (192 MB) -> gathers hit L2.
// ---------------------------------------------------------------------------
__global__ void gcn_aggregate(const float* __restrict__ x,
                              const float* __restrict__ dinv,
                              const unsigned* __restrict__ mask,
                              float* __restrict__ h) {
  const int i = blockIdx.x;
  const int f = threadIdx.x;
  const float di = dinv[i];
  float acc = di * x[(size_t)i * D_FEAT + f];   // identity contribution
  const unsigned* row = mask + (size_t)i * ROW_WORDS;
  for (int w = 0; w < ROW_WORDS; ++w) {
    unsigned word = row[w];
    while (word) {
      const int b = __builtin_ctz(word);
      word &= word - 1;
      const int j = (w << 5) + b;
      acc = fmaf(dinv[j], x[(size_t)j * D_FEAT + f], acc);
    }
  }
  h[(size_t)i * D_FEAT + f] = di * acc;
}

// ---------------------------------------------------------------------------
// K4: out = h @ W (fp32) with V_WMMA_F32_16X16X4_F32.
//     512 threads = 16 waves per block; one block per 16-row m-tile.
//     A-tile h[m0:m0+16, :] staged once in LDS (TDM tensor_load_to_lds with
//     pad fields producing a [16][260] layout -> conflict-free b64 A reads),
//     shared by all 16 waves (one 16x16 n-tile each). B read from global
//     (each element used once per wave; W is L2-resident).
//     A (16x4 f32): lane L holds M=L&15, K={2*(L>>4), 2*(L>>4)+1}
//     C/D (16x16 f32): VGPR r, lanes 0-15 -> M=r, lanes 16-31 -> M=r+8
// ---------------------------------------------------------------------------
__global__ __launch_bounds__(512)
void gcn_gemm_wmma(const float* __restrict__ h,
                   const float* __restrict__ w,
                   float* __restrict__ out) {
  __shared__ __align__(16) float As[16][APAD];   // 16,640 B

  const int m0 = blockIdx.x * 16;

#if __has_builtin(__builtin_amdgcn_tensor_load_to_lds)
  // --- Stage A-tile with the Tensor Data Mover (one wave issues the DMA) ---
  if (threadIdx.x < 32) {
    const unsigned long long ga = (unsigned long long)(uintptr_t)(h + (size_t)m0 * D_FEAT);
    const unsigned lds_off = (unsigned)(uintptr_t)(&As[0][0]);  // low 32 bits = LDS offset
    v4u g0;
    g0.x = 1u;                                   // count=1, user mode, no gather
    g0.y = lds_off;                              // lds_addr
    g0.z = (unsigned)ga;                         // global_addr[31:0]
    g0.w = (unsigned)(ga >> 32) | (2u << 30);    // global_addr[56:32] | type=2
    v8i g1;
    g1[0] = (2 << 16)        // data_size = 4 bytes
          | (1 << 20)        // pad_enable
          | (7 << 22)        // pad_interval: every 256 DWORDs
          | (3 << 25);       // pad_amount: 4 DWORDs  -> row stride 260 floats
    g1[1] = (int)(256u << 16);     // tensor_dim0 = 256 (bits 63:48)
    g1[2] = (int)(10000u << 16);   // tensor_dim0 hi=0; tensor_dim1 = 10000 (bits 95:80)
    g1[3] = (int)(256u << 16);     // tensor_dim1 hi=0; tile_dim0 = 256 (bits 127:112)
    g1[4] = 16;                    // tile_dim1 = 16, tile_dim2 = 0
    g1[5] = 256;                   // tensor_dim0_stride = 256 (row stride of h)
    g1[6] = 0;
    g1[7] = 0;
    const v4i z4 = {0, 0, 0, 0};
#if defined(__clang_major__) && (__clang_major__ >= 23)
    const v8i z8 = {0, 0, 0, 0, 0, 0, 0, 0};
    __builtin_amdgcn_tensor_load_to_lds(g0, g1, z4, z4, z8, 0);
#else
    __builtin_amdgcn_tensor_load_to_lds(g0, g1, z4, z4, 0);
#endif
    __builtin_amdgcn_s_wait_tensorcnt(0);
  }
#else
  // --- Fallback: cooperative staging, 8 floats per thread -----------------
  {
    const int m = threadIdx.x >> 5;
    const int k = (threadIdx.x & 31) * 8;
    const float4* src = (const float4*)(h + (size_t)(m0 + m) * D_FEAT + k);
    const float4 a0 = src[0];
    const float4 a1 = src[1];
    *(float4*)&As[m][k]     = a0;
    *(float4*)&As[m][k + 4] = a1;
  }
#endif
  __syncthreads();

  const int lane = threadIdx.x & 31;
  const int n0   = (threadIdx.x >> 5) * 16;   // wave id -> n-tile
  const int half = lane >> 4;                 // selects K pair {0,1} vs {2,3}
  const int lid  = lane & 15;

  v8f c = {};
#pragma unroll 4
  for (int k = 0; k < D_FEAT; k += 4) {
    const int kk = k + 2 * half;
    v2f a = *(const v2f*)(&As[lid][kk]);      // LDS, conflict-free b64
    v2f b;
    b.x = w[(size_t)(kk + 0) * D_FEAT + n0 + lid];
    b.y = w[(size_t)(kk + 1) * D_FEAT + n0 + lid];
    c = __builtin_amdgcn_wmma_f32_16x16x4_f32(
            /*neg_a=*/false, a, /*neg_b=*/false, b,
            /*c_mod=*/(short)0, c, /*reuse_a=*/false, /*reuse_b=*/false);
  }
#pragma unroll
  for (int r = 0; r < 8; ++r) {
    out[(size_t)(m0 + r + 8 * half) * D_FEAT + n0 + lid] = c[r];
  }
}

// ---------------------------------------------------------------------------
extern "C" void kernel_launch(void* const* d_in, const int* in_sizes, int n_in,
                              void* d_out, int out_size, void* d_ws, size_t ws_size,
                              hipStream_t stream) {
  const float* x   = (const float*)d_in[0];   // [N, 256] f32
  const int*   ei  = (const int*)d_in[1];     // [2, E] (row0=src, row1=dst)
  const float* wt  = (const float*)d_in[2];   // [256, 256] f32
  float*       out = (float*)d_out;           // [N, 256] f32
  const int E = in_sizes[1] / 2;

  char* ws = (char*)d_ws;
  unsigned* mask = (unsigned*)ws;                        // 12,520,000 B
  float*    dinv = (float*)(ws + MASK_BYTES);            //     40,000 B
  float*    h    = (float*)(ws + MASK_BYTES + 40000);    // 10,240,000 B

  gcn_zero<<<2048, 256, 0, stream>>>((uint4*)mask, (long)(MASK_BYTES / 16));
  gcn_build_adj<<<(E + 255) / 256, 256, 0, stream>>>(ei, E, mask);
  gcn_degree<<<(N_NODES + 255) / 256, 256, 0, stream>>>(mask, dinv);
  gcn_aggregate<<<N_NODES, 256, 0, stream>>>(x, dinv, mask, h);
  // one block per m-tile row: 625 blocks x 512 threads (16 waves = 16 n-tiles)
  gcn_gemm_wmma<<<N_NODES / 16, 512, 0, stream>>>(h, wt, out);
}